// LJ126_65077344469604
// MI455X (gfx1250) — compile-verified
//
#include <hip/hip_runtime.h>
#include <stdint.h>

#define TPB 256

// Relaxed agent-scope float atomic add -> global_atomic_add_f32 (verified by vmem count).
__device__ __forceinline__ void atomic_add_f32(float* p, float v) {
    __hip_atomic_fetch_add(p, v, __ATOMIC_RELAXED, __HIP_MEMORY_SCOPE_AGENT);
}

// CDNA5 async global->LDS copy (ASYNCcnt-tracked). lds_byte_addr = offset within
// the workgroup LDS allocation (low 32 bits of a generic pointer to __shared__).
__device__ __forceinline__ void async_load_b128_to_lds(uint32_t lds_byte_addr,
                                                       const void* gaddr) {
    asm volatile("global_load_async_to_lds_b128 %0, %1, off"
                 :: "v"(lds_byte_addr), "v"(gaddr)
                 : "memory");
}

__device__ __forceinline__ void lj_pair(int i, int j,
                                        const float* __restrict__ R,
                                        float sig2, float cut2, float eps,
                                        float* __restrict__ energy,
                                        float* __restrict__ forces) {
    float rix = R[3 * i + 0], riy = R[3 * i + 1], riz = R[3 * i + 2];
    float rjx = R[3 * j + 0], rjy = R[3 * j + 1], rjz = R[3 * j + 2];
    float dx = rix - rjx, dy = riy - rjy, dz = riz - rjz;
    float r2 = dx * dx + dy * dy + dz * dz;
    // ~98% of random pairs fail the cutoff -> skip divide + all 4 atomics.
    if (r2 < cut2 && r2 > 1e-10f) {
        float sr2  = sig2 / r2;
        float sr6  = sr2 * sr2 * sr2;
        float sr12 = sr6 * sr6;
        float e    = 2.0f * eps * (sr12 - sr6);              // 0.5 * 4eps folded
        float fm   = 24.0f * eps * (2.0f * sr12 - sr6) / r2;
        atomic_add_f32(&energy[i], e);
        atomic_add_f32(&forces[3 * i + 0], fm * dx);
        atomic_add_f32(&forces[3 * i + 1], fm * dy);
        atomic_add_f32(&forces[3 * i + 2], fm * dz);
    }
}

__global__ __launch_bounds__(TPB)
void lj_pairs_kernel(const float* __restrict__ R,
                     const float* __restrict__ eps_p,
                     const float* __restrict__ sig_p,
                     const float* __restrict__ cut_p,
                     const int*  __restrict__ idx_i,
                     const int*  __restrict__ idx_j,
                     float* __restrict__ energy,
                     float* __restrict__ forces,
                     int n_pairs) {
    // [stage][array(i/j)][thread] staging buffers: 16 KB of the 320 KB WGP LDS.
    __shared__ int4 sbuf[2][2][TPB];

    const float eps  = *eps_p;          // uniform -> s_load
    const float sig  = *sig_p;
    const float cut  = *cut_p;
    const float sig2 = sig * sig;
    const float cut2 = cut * cut;

    const int n4     = n_pairs >> 2;
    const int tid    = threadIdx.x;
    const int gtid   = blockIdx.x * TPB + tid;
    const int stride = gridDim.x * TPB;

    const int4* ii4 = reinterpret_cast<const int4*>(idx_i);
    const int4* jj4 = reinterpret_cast<const int4*>(idx_j);

    // LDS byte offsets of this thread's private staging slots.
    const uint32_t lds_i[2] = { (uint32_t)(uintptr_t)&sbuf[0][0][tid],
                                (uint32_t)(uintptr_t)&sbuf[1][0][tid] };
    const uint32_t lds_j[2] = { (uint32_t)(uintptr_t)&sbuf[0][1][tid],
                                (uint32_t)(uintptr_t)&sbuf[1][1][tid] };

    int g = gtid;
    int stage = 0;
    if (g < n4) {                         // prologue: stage 0 in flight
        async_load_b128_to_lds(lds_i[0], ii4 + g);
        async_load_b128_to_lds(lds_j[0], jj4 + g);
    }
    while (g < n4) {
        const int gn = g + stride;
        if (gn < n4) {                    // issue next stage, then wait for current
            const int ns = stage ^ 1;
            async_load_b128_to_lds(lds_i[ns], ii4 + gn);
            async_load_b128_to_lds(lds_j[ns], jj4 + gn);
            asm volatile("s_wait_asynccnt 0x2" ::: "memory");  // newest 2 may remain
        } else {
            asm volatile("s_wait_asynccnt 0x0" ::: "memory");
        }
        // Each thread reads back only the slots it loaded itself -> no barrier needed.
        int4 iv = sbuf[stage][0][tid];
        int4 jv = sbuf[stage][1][tid];

        lj_pair(iv.x, jv.x, R, sig2, cut2, eps, energy, forces);
        lj_pair(iv.y, jv.y, R, sig2, cut2, eps, energy, forces);
        lj_pair(iv.z, jv.z, R, sig2, cut2, eps, energy, forces);
        lj_pair(iv.w, jv.w, R, sig2, cut2, eps, energy, forces);

        g = gn;
        stage ^= 1;
    }

    // Tail pairs (n_pairs % 4): plain loads, first few global threads.
    const int tail = n_pairs & 3;
    if (gtid < tail) {
        const int p = n4 * 4 + gtid;
        lj_pair(idx_i[p], idx_j[p], R, sig2, cut2, eps, energy, forces);
    }
}

// Zero the output buffer (energy[N] ++ forces[N*3]); harness poisons d_out and
// the atomic scatter above requires zeros each call.
__global__ void lj_zero_kernel(float* __restrict__ out, int n4, int tail) {
    int t = blockIdx.x * blockDim.x + threadIdx.x;
    if (t < n4) {
        float4 z = make_float4(0.f, 0.f, 0.f, 0.f);
        reinterpret_cast<float4*>(out)[t] = z;
    }
    if (t < tail) {
        out[n4 * 4 + t] = 0.f;
    }
}

extern "C" void kernel_launch(void* const* d_in, const int* in_sizes, int n_in,
                              void* d_out, int out_size, void* d_ws, size_t ws_size,
                              hipStream_t stream) {
    const float* R     = (const float*)d_in[0];   // [N*3]
    const float* eps_p = (const float*)d_in[1];   // [1]
    const float* sig_p = (const float*)d_in[2];   // [1]
    const float* cut_p = (const float*)d_in[3];   // [1]
    const int*   idx_i = (const int*)d_in[4];     // [P]
    const int*   idx_j = (const int*)d_in[5];     // [P]

    const int N = in_sizes[0] / 3;
    const int P = in_sizes[4];

    float* energy = (float*)d_out;      // [N]
    float* forces = energy + N;         // [N*3]

    // 1) zero outputs
    int zn4   = out_size >> 2;
    int ztail = out_size & 3;
    int zblk  = (zn4 + TPB - 1) / TPB;
    if (zblk < 1) zblk = 1;
    lj_zero_kernel<<<zblk, TPB, 0, stream>>>((float*)d_out, zn4, ztail);

    // 2) pair scatter: persistent grid, ~8 int4-groups per thread so the
    //    2-stage async-LDS pipeline has work to overlap.
    int n4 = P >> 2;
    long long want = ((long long)n4 + (long long)TPB * 8 - 1) / ((long long)TPB * 8);
    int blk = (int)(want < 1 ? 1 : want);
    lj_pairs_kernel<<<blk, TPB, 0, stream>>>(R, eps_p, sig_p, cut_p,
                                             idx_i, idx_j, energy, forces, P);
}